// AsymmetricLossCustom_18064632447145
// MI455X (gfx1250) — compile-verified
//
#include <hip/hip_runtime.h>
#include <cstdint>

#define B_DIM 4096
#define C_DIM 9605
#define CLIPV 0.05f
#define EPSV 1e-8f
#define G_MAIN 4096
#define G_ROWS 512            // 8 rows (waves) per block
#define NCHUNK 38420          // (4096*9605)/1024 exactly

typedef __attribute__((ext_vector_type(2))) float v2f;
typedef __attribute__((ext_vector_type(8))) float v8f;

// ---------- per-element loss (matches reference op order) ----------
__device__ __forceinline__ float loss1(float xv, float yv) {
    float e   = __expf(-xv);
    float sp  = 1.0f / (1.0f + e);                    // sigmoid
    float lp  = __logf(fmaxf(sp, EPSV));
    float sn  = fminf((1.0f - sp) + CLIPV, 1.0f);
    float ln_ = __logf(fmaxf(sn, EPSV));
    return fmaf(yv, lp - ln_, ln_);                   // y*lp + (1-y)*ln
}

// ---------- CDNA5 async global->LDS (tracked by ASYNCcnt) ----------
__device__ __forceinline__ void async_ld16(uint32_t lds_addr, const void* gaddr) {
    asm volatile("global_load_async_to_lds_b128 %0, %1, off"
                 :: "v"(lds_addr), "v"((unsigned long long)(uintptr_t)gaddr)
                 : "memory");
}

// ---------- block reduction of 256 floats via chained V_WMMA_F32_16X16X4_F32 ----
// B = ones(4x16)  =>  D[m,n] = sum_k A[m,k] + C[m,n]; chaining C accumulates.
// Any fixed column n of the final D holds the 16 row-sums; column n=0 lives in
// lane 0 (m=0..7) and lane 16 (m=8..15). Fixed hardware order => deterministic.
// All 256 threads run the WMMA (no divergence) so EXEC is all-ones per wave.
__device__ __forceinline__ float block_total_wmma(float acc, float* red, int tid) {
    red[tid] = acc;
    __syncthreads();
    const int lane = tid & 31;
    v2f b; b[0] = 1.0f; b[1] = 1.0f;
    v8f d = {};
#pragma unroll
    for (int i = 0; i < 4; ++i) {
        v2f a;
        a[0] = red[64 * i + 2 * lane];
        a[1] = red[64 * i + 2 * lane + 1];
        d = __builtin_amdgcn_wmma_f32_16x16x4_f32(false, a, false, b,
                                                  (short)0, d, false, false);
    }
    float s = d[0] + d[1] + d[2] + d[3] + d[4] + d[5] + d[6] + d[7];
    __syncthreads();
    red[tid] = s;
    __syncthreads();
    return red[0] + red[16];   // column 0 of D: m=0..7 (lane0) + m=8..15 (lane16)
}

// ---------- kernel 1: column masks + dedup flags ----------
__global__ void __launch_bounds__(256)
k_setup(const int* __restrict__ rind, const int* __restrict__ dind,
        const int* __restrict__ cind, unsigned* __restrict__ cmask,
        int* __restrict__ colArr, int* __restrict__ isFirst) {
    int t = threadIdx.x;
    for (int c = t; c < C_DIM; c += 256) cmask[c] = 0u;
    __syncthreads();
    if (t < 170) {
        int col, bit;
        if (t < 70)       { col = rind[t];       bit = 1; }
        else if (t < 140) { col = dind[t - 70];  bit = 2; }
        else              { col = cind[t - 140]; bit = 4; }
        colArr[t] = col;
        atomicOr(&cmask[col], (unsigned)bit);
    }
    __syncthreads();
    if (t < 170) {
        int col = colArr[t];
        int first = 1;
        for (int j = 0; j < t; ++j)
            if (colArr[j] == col) { first = 0; break; }
        isFirst[t] = first;
    }
}

// ---------- kernel 2: main unscaled streaming sum via async LDS pipeline ----------
__global__ void __launch_bounds__(256)
k_main(const float* __restrict__ x, const float* __restrict__ y,
       float* __restrict__ partials) {
    const int tid = threadIdx.x;
    __shared__ float4 sb[2][2][256];   // [buf][x/y][thread] = 16 KB
    __shared__ float red[256];

    uint32_t lx[2], ly[2];
    lx[0] = (uint32_t)(uintptr_t)&sb[0][0][tid];
    ly[0] = (uint32_t)(uintptr_t)&sb[0][1][tid];
    lx[1] = (uint32_t)(uintptr_t)&sb[1][0][tid];
    ly[1] = (uint32_t)(uintptr_t)&sb[1][1][tid];

    const float4* x4 = (const float4*)x;
    const float4* y4 = (const float4*)y;
    const long G = (long)gridDim.x;

    long cur = (long)blockIdx.x;
    // prologue: stage first chunk into buffer 0
    if (cur < NCHUNK) {
        size_t o = (size_t)cur * 256 + tid;
        async_ld16(lx[0], x4 + o);
        async_ld16(ly[0], y4 + o);
    }

    float acc = 0.0f;
    int buf = 0;
    for (long c = cur; c < NCHUNK; c += G) {
        long n = c + G;
        if (n < NCHUNK) {
            size_t o = (size_t)n * 256 + tid;
            async_ld16(lx[buf ^ 1], x4 + o);
            async_ld16(ly[buf ^ 1], y4 + o);
            // oldest pair (current buffer) complete when <=2 remain outstanding
            asm volatile("s_wait_asynccnt 0x2" ::: "memory");
        } else {
            asm volatile("s_wait_asynccnt 0x0" ::: "memory");
        }
        // each lane consumes the slot it loaded: wave-local, no barrier needed
        float4 vx = sb[buf][0][tid];
        float4 vy = sb[buf][1][tid];
        acc += loss1(vx.x, vy.x);
        acc += loss1(vx.y, vy.y);
        acc += loss1(vx.z, vy.z);
        acc += loss1(vx.w, vy.w);
        buf ^= 1;
    }

    float tot = block_total_wmma(acc, red, tid);
    if (tid == 0) partials[blockIdx.x] = -tot;   // negated: final = sum(partials)
}

// ---------- kernel 3: per-row group flags + 0.9x active correction ----------
__global__ void __launch_bounds__(256)
k_rows(const float* __restrict__ x, const float* __restrict__ y,
       const int* __restrict__ rind, const int* __restrict__ dind,
       const int* __restrict__ cind, const unsigned* __restrict__ cmask,
       const int* __restrict__ colArr, const int* __restrict__ isFirst,
       float* __restrict__ partials) {
    const int tid = threadIdx.x;
    const int lane = tid & 31;
    const int w = tid >> 5;
    __shared__ unsigned flags[8];
    __shared__ float red[256];

    const int row = blockIdx.x * 8 + w;
    const float* xr = x + (size_t)row * C_DIM;
    const float* yr = y + (size_t)row * C_DIM;

    if (lane == 0) flags[w] = 0u;
    __syncthreads();

    unsigned f = 0u;
    for (int i = lane; i < 70; i += 32) if (yr[rind[i]] > 0.0f) f |= 1u;
    for (int i = lane; i < 70; i += 32) if (yr[dind[i]] > 0.0f) f |= 2u;
    for (int i = lane; i < 30; i += 32) if (yr[cind[i]] > 0.0f) f |= 4u;
    if (f) atomicOr(&flags[w], f);
    __syncthreads();

    unsigned hb = flags[w];
    unsigned ag = hb ? (~hb & 7u) : 0u;   // active groups: fired elsewhere, not here

    float corr = 0.0f;
    if (ag) {
        for (int e = lane; e < 170; e += 32) {
            if (isFirst[e]) {
                int col = colArr[e];
                if (cmask[col] & ag)
                    corr += 0.9f * loss1(xr[col], yr[col]);
            }
        }
    }

    float tot = block_total_wmma(corr, red, tid);
    if (tid == 0) partials[blockIdx.x] = tot;
}

// ---------- kernel 4: deterministic final reduction (double, fixed order) ------
__global__ void __launch_bounds__(256)
k_final(const float* __restrict__ partials, int n, float* __restrict__ out) {
    __shared__ double red[256];
    int t = threadIdx.x;
    double a = 0.0;
    for (int i = t; i < n; i += 256) a += (double)partials[i];
    red[t] = a;
    __syncthreads();
    for (int s = 128; s > 0; s >>= 1) {
        if (t < s) red[t] += red[t + s];
        __syncthreads();
    }
    if (t == 0) out[0] = (float)red[0];
}

extern "C" void kernel_launch(void* const* d_in, const int* in_sizes, int n_in,
                              void* d_out, int out_size, void* d_ws, size_t ws_size,
                              hipStream_t stream) {
    const float* x    = (const float*)d_in[0];
    const float* y    = (const float*)d_in[1];
    const int*   rind = (const int*)d_in[2];
    const int*   dind = (const int*)d_in[3];
    const int*   cind = (const int*)d_in[4];
    float* out = (float*)d_out;

    char* ws = (char*)d_ws;
    unsigned* cmask  = (unsigned*)(ws);            // 9605 * 4 B (pad to 40960)
    int* colArr      = (int*)(ws + 40960);         // 170 * 4 B (pad to 1024)
    int* isFirst     = (int*)(ws + 41984);         // 170 * 4 B (pad to 1024)
    float* pmain     = (float*)(ws + 43008);       // G_MAIN floats
    float* prows     = pmain + G_MAIN;             // G_ROWS floats
    const int npart  = G_MAIN + G_ROWS;

    k_setup<<<1, 256, 0, stream>>>(rind, dind, cind, cmask, colArr, isFirst);
    k_main<<<G_MAIN, 256, 0, stream>>>(x, y, pmain);
    k_rows<<<G_ROWS, 256, 0, stream>>>(x, y, rind, dind, cind, cmask, colArr,
                                       isFirst, prows);
    k_final<<<1, 256, 0, stream>>>(pmain, npart, out);
}